// BMAM_16492674417424
// MI455X (gfx1250) — compile-verified
//
#include <hip/hip_runtime.h>
#include <hip/hip_bf16.h>

typedef __attribute__((ext_vector_type(2))) float v2f;
typedef __attribute__((ext_vector_type(8))) float v8f;

#define BATCH   8
#define SEQLEN  4096
#define DMODEL  256
#define DINNER  512
#define DSTATE  16
#define DTRANK  16

__device__ __forceinline__ v8f wmma4(v2f a, v2f b, v8f c) {
  // V_WMMA_F32_16X16X4_F32: (neg_a, A, neg_b, B, c_mod, C, reuse_a, reuse_b)
  return __builtin_amdgcn_wmma_f32_16x16x4_f32(false, a, false, b, (short)0, c,
                                               false, false);
}

// ---------------------------------------------------------------------------
// Kernel 1: xz = x @ W_in ; cols [0,512) -> xi, cols [512,1024) -> z
// M=32768, K=256, N=1024. Register-blocked: each wave computes a 64x64 block
// (4 M-tiles x 4 N-tiles, 16 accumulators) so A/B frags are reused 4x each.
// ---------------------------------------------------------------------------
__global__ __launch_bounds__(256) void k_gemm_in(const float* __restrict__ X,
                                                 const float* __restrict__ Win,
                                                 float* __restrict__ xi,
                                                 float* __restrict__ z) {
  const int lane = threadIdx.x & 31;
  const int wave = threadIdx.x >> 5;
  const int job  = blockIdx.x * 8 + wave;      // 512*16 = 8192 jobs
  const int NB   = (2 * DINNER) / 64;          // 16 col blocks
  const int m0   = (job / NB) * 64;
  const int n0   = (job % NB) * 64;
  const int half = lane >> 4;
  const int l15  = lane & 15;
  const int koff = half * 2;

  v8f acc[4][4];
#pragma unroll
  for (int i = 0; i < 4; ++i)
#pragma unroll
    for (int j = 0; j < 4; ++j) acc[i][j] = (v8f){};

  for (int k = 0; k < DMODEL; k += 4) {
    v2f af[4], bf[4];
#pragma unroll
    for (int i = 0; i < 4; ++i)
      af[i] = *(const v2f*)(X + (size_t)(m0 + i * 16 + l15) * DMODEL + k + koff);
#pragma unroll
    for (int j = 0; j < 4; ++j) {
      const int n = n0 + j * 16 + l15;
      bf[j].x = Win[(size_t)(k + koff)     * (2 * DINNER) + n];
      bf[j].y = Win[(size_t)(k + koff + 1) * (2 * DINNER) + n];
    }
#pragma unroll
    for (int i = 0; i < 4; ++i)
#pragma unroll
      for (int j = 0; j < 4; ++j) acc[i][j] = wmma4(af[i], bf[j], acc[i][j]);
  }

#pragma unroll
  for (int i = 0; i < 4; ++i) {
    const int mbase = m0 + i * 16 + half * 8;
#pragma unroll
    for (int j = 0; j < 4; ++j) {
      const int n   = n0 + j * 16 + l15;
      float* dst    = (n < DINNER) ? xi : z;
      const int col = (n < DINNER) ? n : (n - DINNER);
#pragma unroll
      for (int r = 0; r < 8; ++r)
        dst[(size_t)(mbase + r) * DINNER + col] = acc[i][j][r];
    }
  }
}

// ---------------------------------------------------------------------------
// Kernel 2: depthwise causal conv (k=4) + bias + SiLU
// ---------------------------------------------------------------------------
__global__ __launch_bounds__(256) void k_conv(const float* __restrict__ xi,
                                              const float* __restrict__ cw,
                                              const float* __restrict__ cb,
                                              float* __restrict__ xcl) {
  const int idx = blockIdx.x * blockDim.x + threadIdx.x;
  const int d = idx & (DINNER - 1);
  const int l = (idx >> 9) & (SEQLEN - 1);
  const int b = idx >> 21;
  const float* base = xi + (size_t)b * SEQLEN * DINNER + d;
  float acc = cb[d];
#pragma unroll
  for (int j = 0; j < 4; ++j) {
    const int ls = l - 3 + j;
    if (ls >= 0) acc += cw[d * 4 + j] * base[(size_t)ls * DINNER];
  }
  xcl[idx] = acc / (1.f + __expf(-acc));       // SiLU
}

// ---------------------------------------------------------------------------
// Kernel 3: x_dbl = xcl @ W_x -> dtp/Bm/Cm, each [M,16].
// M=32768, K=512, N=48. One wave per M-tile computes all 3 N-tiles so the
// xcl row-panel is read exactly once.
// ---------------------------------------------------------------------------
__global__ __launch_bounds__(256) void k_gemm_xdbl(const float* __restrict__ xcl,
                                                   const float* __restrict__ Wx,
                                                   float* __restrict__ dtp,
                                                   float* __restrict__ Bm,
                                                   float* __restrict__ Cm) {
  const int lane = threadIdx.x & 31;
  const int wave = threadIdx.x >> 5;
  const int tm   = blockIdx.x * 8 + wave;      // 2048 M-tiles
  const int half = lane >> 4;
  const int l15  = lane & 15;
  const int m    = tm * 16 + l15;
  const int koff = half * 2;
  const int NW   = DTRANK + 2 * DSTATE;        // 48

  v8f acc[3];
#pragma unroll
  for (int j = 0; j < 3; ++j) acc[j] = (v8f){};

  const float* arow = xcl + (size_t)m * DINNER;
  for (int k = 0; k < DINNER; k += 4) {
    v2f a = *(const v2f*)(arow + k + koff);
#pragma unroll
    for (int j = 0; j < 3; ++j) {
      const int n = j * 16 + l15;
      v2f b; b.x = Wx[(size_t)(k + koff)     * NW + n];
             b.y = Wx[(size_t)(k + koff + 1) * NW + n];
      acc[j] = wmma4(a, b, acc[j]);
    }
  }

  const int mbase = tm * 16 + half * 8;
#pragma unroll
  for (int j = 0; j < 3; ++j) {
    float* dst = (j == 0) ? dtp : ((j == 1) ? Bm : Cm);
#pragma unroll
    for (int r = 0; r < 8; ++r)
      dst[(size_t)(mbase + r) * 16 + l15] = acc[j][r];
  }
}

// ---------------------------------------------------------------------------
// Kernel 4: delta = softplus(dtp @ W_dt + b_dt)   M=32768, K=16, N=512.
// 1 M-tile x 4 N-tiles per wave (A panel read once per 64 output cols).
// ---------------------------------------------------------------------------
__global__ __launch_bounds__(256) void k_gemm_delta(const float* __restrict__ dtp,
                                                    const float* __restrict__ Wdt,
                                                    const float* __restrict__ bdt,
                                                    float* __restrict__ delta) {
  const int lane = threadIdx.x & 31;
  const int wave = threadIdx.x >> 5;
  const int job  = blockIdx.x * 8 + wave;      // 2048 * 8 = 16384 jobs
  const int NB   = DINNER / 64;                // 8 col blocks
  const int tm   = job / NB;
  const int n0   = (job % NB) * 64;
  const int half = lane >> 4;
  const int l15  = lane & 15;
  const int m    = tm * 16 + l15;
  const int koff = half * 2;

  v8f acc[4];
#pragma unroll
  for (int j = 0; j < 4; ++j) acc[j] = (v8f){};

  const float* arow = dtp + (size_t)m * DTRANK;
#pragma unroll
  for (int k = 0; k < DTRANK; k += 4) {
    v2f a = *(const v2f*)(arow + k + koff);
#pragma unroll
    for (int j = 0; j < 4; ++j) {
      const int n = n0 + j * 16 + l15;
      v2f b; b.x = Wdt[(size_t)(k + koff)     * DINNER + n];
             b.y = Wdt[(size_t)(k + koff + 1) * DINNER + n];
      acc[j] = wmma4(a, b, acc[j]);
    }
  }

  const int mbase = tm * 16 + half * 8;
#pragma unroll
  for (int j = 0; j < 4; ++j) {
    const int n = n0 + j * 16 + l15;
    const float bv = bdt[n];
#pragma unroll
    for (int r = 0; r < 8; ++r) {
      const float v  = acc[j][r] + bv;
      const float sp = (v > 20.f) ? v : log1pf(__expf(v));   // softplus
      delta[(size_t)(mbase + r) * DINNER + n] = sp;
    }
  }
}

// ---------------------------------------------------------------------------
// Kernel 5: selective scan. One lane per (b,d,n); 16-lane groups reduce y over
// n with __shfl_xor (wave32). Epilogue fused: yg = (y + x*D) * silu(z).
// ---------------------------------------------------------------------------
__global__ __launch_bounds__(256) void k_scan(const float* __restrict__ xcl,
                                              const float* __restrict__ delta,
                                              const float* __restrict__ Bm,
                                              const float* __restrict__ Cm,
                                              const float* __restrict__ z,
                                              const float* __restrict__ Alog,
                                              const float* __restrict__ Dv,
                                              float* __restrict__ yg) {
  const int gid = blockIdx.x * blockDim.x + threadIdx.x;
  const int n = gid & (DSTATE - 1);
  const int d = (gid >> 4) & (DINNER - 1);
  const int b = gid >> 13;

  const float An = -__expf(Alog[d * DSTATE + n]);
  const float Dd = Dv[d];

  const float* xp = xcl   + (size_t)b * SEQLEN * DINNER + d;
  const float* dp = delta + (size_t)b * SEQLEN * DINNER + d;
  const float* zp = z     + (size_t)b * SEQLEN * DINNER + d;
  const float* bp = Bm    + (size_t)b * SEQLEN * DSTATE + n;
  const float* cp = Cm    + (size_t)b * SEQLEN * DSTATE + n;
  float*       yp = yg    + (size_t)b * SEQLEN * DINNER + d;

  float h = 0.f;
  for (int t = 0; t < SEQLEN; ++t) {
    const float xt  = xp[(size_t)t * DINNER];
    const float dtv = dp[(size_t)t * DINNER];
    const float bt  = bp[(size_t)t * DSTATE];
    const float ct  = cp[(size_t)t * DSTATE];
    h = __expf(dtv * An) * h + (dtv * bt) * xt;
    float yn = h * ct;
    yn += __shfl_xor(yn, 1);
    yn += __shfl_xor(yn, 2);
    yn += __shfl_xor(yn, 4);
    yn += __shfl_xor(yn, 8);
    if (n == 0) {
      const float zv   = zp[(size_t)t * DINNER];
      const float gate = zv / (1.f + __expf(-zv));
      yp[(size_t)t * DINNER] = (yn + xt * Dd) * gate;   // aliases z: read-before-write
    }
  }
}

// ---------------------------------------------------------------------------
// Kernel 6: out[b,c,l] = sum_d yg[b,l,d] * W_out[d,c]   (transposed store)
// M=32768, K=512, N=256. Register-blocked 4 M-tiles x 4 N-tiles per wave.
// ---------------------------------------------------------------------------
__global__ __launch_bounds__(256) void k_gemm_out(const float* __restrict__ yg,
                                                  const float* __restrict__ Wout,
                                                  float* __restrict__ out) {
  const int lane = threadIdx.x & 31;
  const int wave = threadIdx.x >> 5;
  const int job  = blockIdx.x * 8 + wave;      // 512*4 = 2048 jobs
  const int NB   = DMODEL / 64;                // 4 col blocks
  const int m0   = (job / NB) * 64;
  const int n0   = (job % NB) * 64;
  const int half = lane >> 4;
  const int l15  = lane & 15;
  const int koff = half * 2;

  v8f acc[4][4];
#pragma unroll
  for (int i = 0; i < 4; ++i)
#pragma unroll
    for (int j = 0; j < 4; ++j) acc[i][j] = (v8f){};

  for (int k = 0; k < DINNER; k += 4) {
    v2f af[4], bf[4];
#pragma unroll
    for (int i = 0; i < 4; ++i)
      af[i] = *(const v2f*)(yg + (size_t)(m0 + i * 16 + l15) * DINNER + k + koff);
#pragma unroll
    for (int j = 0; j < 4; ++j) {
      const int n = n0 + j * 16 + l15;
      bf[j].x = Wout[(size_t)(k + koff)     * DMODEL + n];
      bf[j].y = Wout[(size_t)(k + koff + 1) * DMODEL + n];
    }
#pragma unroll
    for (int i = 0; i < 4; ++i)
#pragma unroll
      for (int j = 0; j < 4; ++j) acc[i][j] = wmma4(af[i], bf[j], acc[i][j]);
  }

#pragma unroll
  for (int i = 0; i < 4; ++i) {
    const int mbase = m0 + i * 16 + half * 8;
#pragma unroll
    for (int j = 0; j < 4; ++j) {
      const int n = n0 + j * 16 + l15;
#pragma unroll
      for (int r = 0; r < 8; ++r) {
        const int row = mbase + r;             // row = b*SEQLEN + l
        const int bo  = row >> 12;
        const int lo  = row & (SEQLEN - 1);
        out[(size_t)bo * DMODEL * SEQLEN + (size_t)n * SEQLEN + lo] = acc[i][j][r];
      }
    }
  }
}

// ---------------------------------------------------------------------------
extern "C" void kernel_launch(void* const* d_in, const int* in_sizes, int n_in,
                              void* d_out, int out_size, void* d_ws, size_t ws_size,
                              hipStream_t stream) {
  const float* X    = (const float*)d_in[0];
  const float* Win  = (const float*)d_in[1];
  const float* cw   = (const float*)d_in[2];
  const float* cb   = (const float*)d_in[3];
  const float* Wx   = (const float*)d_in[4];
  const float* Wdt  = (const float*)d_in[5];
  const float* bdt  = (const float*)d_in[6];
  const float* Alog = (const float*)d_in[7];
  const float* Dv   = (const float*)d_in[8];
  const float* Wout = (const float*)d_in[9];
  float* out = (float*)d_out;

  char* ws = (char*)d_ws;
  const size_t SZ_BLD = (size_t)BATCH * SEQLEN * DINNER * sizeof(float);   // 64 MiB
  const size_t SZ_BLN = (size_t)BATCH * SEQLEN * DSTATE * sizeof(float);   //  2 MiB
  float* xi    = (float*)(ws);
  float* z     = (float*)(ws + SZ_BLD);
  float* xcl   = (float*)(ws + 2 * SZ_BLD);
  float* dtp   = (float*)(ws + 3 * SZ_BLD);
  float* Bm    = (float*)(ws + 3 * SZ_BLD + SZ_BLN);
  float* Cm    = (float*)(ws + 3 * SZ_BLD + 2 * SZ_BLN);
  float* delta = xi;   // xi dead after conv
  float* yg    = z;    // scan reads z[t] then overwrites same element

  (void)in_sizes; (void)n_in; (void)out_size; (void)ws_size;

  // xz GEMM: 8192 wave-jobs (64x64 each), 8 per block
  k_gemm_in  <<<1024, 256, 0, stream>>>(X, Win, xi, z);
  // conv+SiLU: 16.7M elements
  k_conv     <<<(BATCH * SEQLEN * DINNER) / 256, 256, 0, stream>>>(xi, cw, cb, xcl);
  // x_dbl GEMM: 2048 M-tiles, 3 N-tiles each
  k_gemm_xdbl<<<256, 256, 0, stream>>>(xcl, Wx, dtp, Bm, Cm);
  // delta GEMM: 16384 wave-jobs (16x64 each)
  k_gemm_delta<<<2048, 256, 0, stream>>>(dtp, Wdt, bdt, delta);
  // selective scan: 65536 lanes
  k_scan     <<<256, 256, 0, stream>>>(xcl, delta, Bm, Cm, z, Alog, Dv, yg);
  // output GEMM + transpose: 2048 wave-jobs (64x64 each)
  k_gemm_out <<<256, 256, 0, stream>>>(yg, Wout, out);
}